// GINAttNet_54065048322603
// MI455X (gfx1250) — compile-verified
//
#include <hip/hip_runtime.h>
#include <hip/hip_bf16.h>
#include <math.h>

#define F_IN      128
#define DIM       95
#define DIMP      96      // padded feature stride (zero col 95)
#define G_BATCH   256
#define DIM5      12
#define NEG_SLOPE 0.2f
#define BN_EPS    1e-5f
#define MTILES    8       // 16-row M tiles per block
#define GEMM_THREADS 192  // 6 waves

typedef float v2f __attribute__((ext_vector_type(2)));
typedef float v8f __attribute__((ext_vector_type(8)));

// ---------------------------------------------------------------- fill
__global__ void fill_kernel(float* __restrict__ p, float v, long long n) {
    long long i = (long long)blockIdx.x * blockDim.x + threadIdx.x;
    long long stride = (long long)gridDim.x * blockDim.x;
    for (; i < n; i += stride) p[i] = v;
}

// ---------------------------------------------------------------- GIN aggregate: agg[dst] += x[src]
// one wave per edge, 32 lanes x float4 = 128 features
__global__ void scatter_x_kernel(const float* __restrict__ x, const int* __restrict__ src,
                                 const int* __restrict__ dst, float* __restrict__ agg, int E) {
    long long wid = ((long long)blockIdx.x * blockDim.x + threadIdx.x) >> 5;
    int lane = threadIdx.x & 31;
    if (wid >= E) return;
    int s = src[wid], d = dst[wid];
    const float4 v = *(const float4*)(x + (size_t)s * F_IN + lane * 4);
    float* o = agg + (size_t)d * F_IN + lane * 4;
    atomicAdd(o + 0, v.x);
    atomicAdd(o + 1, v.y);
    atomicAdd(o + 2, v.z);
    atomicAdd(o + 3, v.w);
}

// ---------------------------------------------------------------- fp32 WMMA GEMM, C[M x 96] = A[M x K] * B[Krows x DIM]
// MODE 0: A = x + agg (stride 128), epilogue bias+ReLU            (GIN layer 1)
// MODE 1: A = buf (stride 96),     epilogue bias+ReLU+BatchNorm   (GIN layer 2 + BN)
// MODE 2: A = buf (stride 96),     epilogue none                  (GAT projection)
// Block = 192 threads = 6 waves; wave w owns cols [16w, 16w+16); block covers MTILES 16-row tiles.
// Both operands staged in LDS with row stride K+4 DWORDs -> conflict-free ds_load_b64 operand fetch.
template <int MODE>
__global__ __launch_bounds__(GEMM_THREADS)
void gemm_wmma_kernel(const float* __restrict__ A0, const float* __restrict__ A1,
                      const float* __restrict__ B, int Krows,
                      const float* __restrict__ bias,
                      const float* __restrict__ bng, const float* __restrict__ bnb,
                      const float* __restrict__ bnm, const float* __restrict__ bnv,
                      float* __restrict__ C, int ntiles) {
    constexpr int KD   = (MODE == 0) ? F_IN : DIMP;  // K extent (padded, multiple of 4)
    constexpr int LSTR = KD + 4;                     // LDS row stride (DWORDs), bank-decorrelating
    constexpr int VPR  = KD / 4;                     // float4 chunks per row

    extern __shared__ float smem[];
    float* Bs = smem;               // B transposed: [DIMP][LSTR]   Bs[n*LSTR + k]
    float* As = smem + DIMP * LSTR; // A tiles:      [MTILES*16][LSTR]

    const int tid   = threadIdx.x;
    const int tile0 = blockIdx.x * MTILES;
    const int valid = min(MTILES, ntiles - tile0);   // uniform per block
    const int row0  = tile0 * 16;

    // ---- stage B transposed (guards applied here, not in the hot loop) ----
    for (int j = tid; j < KD * DIMP; j += GEMM_THREADS) {
        int k = j / DIMP, c = j % DIMP;              // coalesced global read
        float v = (k < Krows && c < DIM) ? B[(size_t)k * DIM + c] : 0.f;
        Bs[c * LSTR + k] = v;
    }
    // ---- stage A tiles (fused x+agg for MODE 0), float4 coalesced ----
    const int nvec = valid * 16 * VPR;
    for (int j = tid; j < nvec; j += GEMM_THREADS) {
        int r  = j / VPR;
        int kc = (j % VPR) * 4;
        float4 v;
        if (MODE == 0) {
            size_t base = (size_t)(row0 + r) * F_IN + kc;
            float4 xa = *(const float4*)(A0 + base);
            float4 ag = *(const float4*)(A1 + base);
            v = make_float4(xa.x + ag.x, xa.y + ag.y, xa.z + ag.z, xa.w + ag.w);
        } else {
            v = *(const float4*)(A0 + (size_t)(row0 + r) * DIMP + kc);
        }
        *(float4*)(As + r * LSTR + kc) = v;
    }
    __syncthreads();

    const int wave = tid >> 5;
    const int lane = tid & 31;
    const int half = lane >> 4;        // 0: lanes 0-15, 1: lanes 16-31
    const int l15  = lane & 15;
    const int ncol = wave * 16 + l15;  // output column this lane owns

    // per-column epilogue constants (uniform across mt loop)
    const int c = (ncol < DIM) ? ncol : 0;
    float bia = 0.f, scale = 0.f, shift = 0.f, mean = 0.f;
    if (MODE != 2) bia = bias[c];
    if (MODE == 1) {
        scale = rsqrtf(bnv[c] + BN_EPS) * bng[c];
        shift = bnb[c];
        mean  = bnm[c];
    }

    const float* Brow = Bs + ncol * LSTR + 2 * half;
    for (int mt = 0; mt < valid; ++mt) {
        const float* Arow = As + (mt * 16 + l15) * LSTR + 2 * half;
        v8f acc = {0.f, 0.f, 0.f, 0.f, 0.f, 0.f, 0.f, 0.f};
#pragma unroll 8
        for (int k0 = 0; k0 < KD; k0 += 4) {
            v2f a = *(const v2f*)(Arow + k0);   // ds_load_b64, conflict-free
            v2f b = *(const v2f*)(Brow + k0);   // ds_load_b64, conflict-free
            acc = __builtin_amdgcn_wmma_f32_16x16x4_f32(
                false, a, false, b, (short)0, acc, false, false);
        }
        const int orow0 = row0 + mt * 16 + 8 * half;
#pragma unroll
        for (int r = 0; r < 8; ++r) {
            float v = acc[r];
            if (MODE == 0) {
                v = fmaxf(v + bia, 0.f);
            } else if (MODE == 1) {
                v = fmaxf(v + bia, 0.f);
                v = (v - mean) * scale + shift;
            }
            if (ncol >= DIM) v = 0.f;            // keep pad column zero
            C[(size_t)(orow0 + r) * DIMP + ncol] = v;
        }
    }
}

// ---------------------------------------------------------------- attention logits per node
__global__ void att_scores_kernel(const float* __restrict__ ht, const float* __restrict__ a_src,
                                  const float* __restrict__ a_dst,
                                  float* __restrict__ als, float* __restrict__ ald, int n) {
    int i = blockIdx.x * blockDim.x + threadIdx.x;
    if (i >= n) return;
    const float* r = ht + (size_t)i * DIMP;
    float s = 0.f, d = 0.f;
    for (int k = 0; k < DIM; ++k) {
        float v = r[k];
        s += v * a_src[k];
        d += v * a_dst[k];
    }
    als[i] = s;
    ald[i] = d;
}

// float atomic max via sign-split int trick (lowers to global_atomic_max/min)
__device__ __forceinline__ void atomicMaxF(float* addr, float val) {
    if (val >= 0.f) atomicMax((int*)addr, __float_as_int(val));
    else            atomicMin((unsigned int*)addr, __float_as_uint(val));
}

__device__ __forceinline__ float edge_score(const float* als, const float* ald, int s, int d) {
    float e = als[s] + ald[d];
    return e > 0.f ? e : NEG_SLOPE * e;
}

// ---------------------------------------------------------------- edge softmax pass 1: segment max
__global__ void edge_max_kernel(const int* __restrict__ src, const int* __restrict__ dst,
                                const float* __restrict__ als, const float* __restrict__ ald,
                                float* __restrict__ mx, int E, int n) {
    long long i = (long long)blockIdx.x * blockDim.x + threadIdx.x;
    if (i >= (long long)E + n) return;
    int s, d;
    if (i < E) { s = src[i]; d = dst[i]; } else { s = d = (int)(i - E); }
    atomicMaxF(mx + d, edge_score(als, ald, s, d));
}

// ---------------------------------------------------------------- pass 2: segment sum of exp
__global__ void edge_den_kernel(const int* __restrict__ src, const int* __restrict__ dst,
                                const float* __restrict__ als, const float* __restrict__ ald,
                                const float* __restrict__ mx, float* __restrict__ den, int E, int n) {
    long long i = (long long)blockIdx.x * blockDim.x + threadIdx.x;
    if (i >= (long long)E + n) return;
    int s, d;
    if (i < E) { s = src[i]; d = dst[i]; } else { s = d = (int)(i - E); }
    float e = edge_score(als, ald, s, d);
    atomicAdd(den + d, expf(e - mx[d]));
}

// ---------------------------------------------------------------- pass 3: weighted feature scatter (wave per edge)
__global__ void edge_agg_kernel(const int* __restrict__ src, const int* __restrict__ dst,
                                const float* __restrict__ als, const float* __restrict__ ald,
                                const float* __restrict__ mx, const float* __restrict__ den,
                                const float* __restrict__ ht, float* __restrict__ outb, int E, int n) {
    long long wid = ((long long)blockIdx.x * blockDim.x + threadIdx.x) >> 5;
    int lane = threadIdx.x & 31;
    if (wid >= (long long)E + n) return;
    int s, d;
    if (wid < E) { s = src[wid]; d = dst[wid]; } else { s = d = (int)(wid - E); }
    float e = edge_score(als, ald, s, d);
    float w = expf(e - mx[d]) / den[d];
    const float* hr = ht + (size_t)s * DIMP;     // pad col is 0 -> harmless
    float* orow = outb + (size_t)d * DIMP;
    int f = lane * 3;                            // 32 lanes x 3 = 96 cols
    atomicAdd(orow + f + 0, w * hr[f + 0]);
    atomicAdd(orow + f + 1, w * hr[f + 1]);
    atomicAdd(orow + f + 2, w * hr[f + 2]);
}

// ---------------------------------------------------------------- global add pool (+ bg bias per node)
__global__ void pool_kernel(const float* __restrict__ outb, const int* __restrict__ batch,
                            const float* __restrict__ bg, float* __restrict__ pool, int n) {
    long long wid = ((long long)blockIdx.x * blockDim.x + threadIdx.x) >> 5;
    int lane = threadIdx.x & 31;
    if (wid >= n) return;
    int b = batch[wid];
    const float* orow = outb + (size_t)wid * DIMP;
    int f = lane * 3;
#pragma unroll
    for (int j = 0; j < 3; ++j) {
        int c = f + j;
        if (c < DIM) atomicAdd(pool + (size_t)b * DIMP + c, orow[c] + bg[c]);
    }
}

// ---------------------------------------------------------------- fc1: g2[256 x 190] = relu(pool @ wfc + bfc)
__global__ void fc_kernel(const float* __restrict__ pool, const float* __restrict__ wfc,
                          const float* __restrict__ bfc, float* __restrict__ g2) {
    int o  = blockIdx.x;    // 0..2*DIM-1
    int gi = threadIdx.x;   // 0..255
    const float* pr = pool + (size_t)gi * DIMP;
    float s = bfc[o];
    for (int k = 0; k < DIM; ++k) s += pr[k] * wfc[(size_t)k * (2 * DIM) + o];
    g2[(size_t)gi * (2 * DIM) + o] = fmaxf(s, 0.f);
}

// ---------------------------------------------------------------- three tiny heads
__global__ void heads_kernel(const float* __restrict__ g2,
                             const float* w10, const float* b10, const float* w20, const float* b20,
                             const float* w11, const float* b11, const float* w21, const float* b21,
                             const float* w12, const float* b12, const float* w22, const float* b22,
                             float* __restrict__ out) {
    int gi = threadIdx.x;   // 0..255, one graph per thread
    const float* gr = g2 + (size_t)gi * (2 * DIM);
    const float* W1[3] = {w10, w11, w12};
    const float* B1[3] = {b10, b11, b12};
    const float* W2[3] = {w20, w21, w22};
    const float* B2[3] = {b20, b21, b22};
    for (int h = 0; h < 3; ++h) {
        float o = B2[h][0];
        for (int j = 0; j < DIM5; ++j) {
            float s = B1[h][j];
            for (int k = 0; k < 2 * DIM; ++k) s += gr[k] * W1[h][(size_t)k * DIM5 + j];
            o += fmaxf(s, 0.f) * W2[h][j];
        }
        if (h == 0) o = 1.f / (1.f + expf(-o));
        out[(size_t)gi * 3 + h] = o;
    }
}

// ================================================================ launcher
extern "C" void kernel_launch(void* const* d_in, const int* in_sizes, int n_in,
                              void* d_out, int out_size, void* d_ws, size_t ws_size,
                              hipStream_t stream) {
    const float* x    = (const float*)d_in[0];
    const int*   ei   = (const int*)d_in[1];
    const int*   batch= (const int*)d_in[2];
    const float* w1a  = (const float*)d_in[3];
    const float* b1a  = (const float*)d_in[4];
    const float* w1b  = (const float*)d_in[5];
    const float* b1b  = (const float*)d_in[6];
    const float* bn_g = (const float*)d_in[7];
    const float* bn_b = (const float*)d_in[8];
    const float* bn_m = (const float*)d_in[9];
    const float* bn_v = (const float*)d_in[10];
    const float* wg   = (const float*)d_in[11];
    const float* bg   = (const float*)d_in[12];
    const float* a_src= (const float*)d_in[13];
    const float* a_dst= (const float*)d_in[14];
    const float* wfc  = (const float*)d_in[15];
    const float* bfc  = (const float*)d_in[16];

    const int n = in_sizes[0] / F_IN;
    const int E = in_sizes[1] / 2;
    const int* src = ei;
    const int* dst = ei + E;

    // workspace layout (floats)
    float* ws = (float*)d_ws;
    size_t off = 0;
    float* agg = ws + off; off += (size_t)n * F_IN;   // reused as GAT output buffer
    float* h1  = ws + off; off += (size_t)n * DIMP;   // reused as ht
    float* h2  = ws + off; off += (size_t)n * DIMP;
    float* als = ws + off; off += n;
    float* ald = ws + off; off += n;
    float* mx  = ws + off; off += n;
    float* den = ws + off; off += n;
    float* pool= ws + off; off += (size_t)G_BATCH * DIMP;
    float* g2  = ws + off; off += (size_t)G_BATCH * 2 * DIM;
    float* outb = agg;  // reuse: agg is dead after GEMM1
    float* ht   = h1;   // reuse: h1 is dead after GEMM2

    const long long T = (long long)E + n;
    const int ntiles  = n / 16;                         // n = 100000 -> 6250 exact
    const int gblocks = (ntiles + MTILES - 1) / MTILES;
    const size_t shmem0 = (size_t)(DIMP + MTILES * 16) * (F_IN + 4) * sizeof(float); // ~115.5 KB
    const size_t shmem1 = (size_t)(DIMP + MTILES * 16) * (DIMP + 4) * sizeof(float); // ~87.5 KB

    // --- GIN aggregate ---
    fill_kernel<<<2048, 256, 0, stream>>>(agg, 0.f, (long long)n * F_IN);
    scatter_x_kernel<<<(unsigned)(((long long)E * 32 + 255) / 256), 256, 0, stream>>>(x, src, dst, agg, E);

    // --- GIN MLP layer 1: h1 = relu((x+agg) @ w1a + b1a) ---
    gemm_wmma_kernel<0><<<gblocks, GEMM_THREADS, shmem0, stream>>>(
        x, agg, w1a, F_IN, b1a, nullptr, nullptr, nullptr, nullptr, h1, ntiles);
    // --- GIN MLP layer 2 + outer ReLU + BatchNorm: h2 ---
    gemm_wmma_kernel<1><<<gblocks, GEMM_THREADS, shmem1, stream>>>(
        h1, nullptr, w1b, DIM, b1b, bn_g, bn_b, bn_m, bn_v, h2, ntiles);
    // --- GAT projection: ht = h2 @ wg ---
    gemm_wmma_kernel<2><<<gblocks, GEMM_THREADS, shmem1, stream>>>(
        h2, nullptr, wg, DIM, nullptr, nullptr, nullptr, nullptr, nullptr, ht, ntiles);

    // --- attention logits ---
    att_scores_kernel<<<(n + 255) / 256, 256, 0, stream>>>(ht, a_src, a_dst, als, ald, n);

    // --- edge softmax + aggregation ---
    fill_kernel<<<512, 256, 0, stream>>>(mx, -__builtin_huge_valf(), n);
    fill_kernel<<<512, 256, 0, stream>>>(den, 0.f, n);
    fill_kernel<<<2048, 256, 0, stream>>>(outb, 0.f, (long long)n * DIMP);
    fill_kernel<<<64, 256, 0, stream>>>(pool, 0.f, (long long)G_BATCH * DIMP);

    edge_max_kernel<<<(unsigned)((T + 255) / 256), 256, 0, stream>>>(src, dst, als, ald, mx, E, n);
    edge_den_kernel<<<(unsigned)((T + 255) / 256), 256, 0, stream>>>(src, dst, als, ald, mx, den, E, n);
    edge_agg_kernel<<<(unsigned)((T * 32 + 255) / 256), 256, 0, stream>>>(src, dst, als, ald, mx, den,
                                                                          ht, outb, E, n);
    // --- pool + heads ---
    pool_kernel<<<(unsigned)(((long long)n * 32 + 255) / 256), 256, 0, stream>>>(outb, batch, bg, pool, n);
    fc_kernel<<<2 * DIM, G_BATCH, 0, stream>>>(pool, wfc, bfc, g2);
    heads_kernel<<<1, G_BATCH, 0, stream>>>(g2,
        (const float*)d_in[17], (const float*)d_in[18], (const float*)d_in[19], (const float*)d_in[20],
        (const float*)d_in[21], (const float*)d_in[22], (const float*)d_in[23], (const float*)d_in[24],
        (const float*)d_in[25], (const float*)d_in[26], (const float*)d_in[27], (const float*)d_in[28],
        (float*)d_out);
    (void)n_in; (void)out_size; (void)ws_size;
}